// AlternateLayer_73607149519122
// MI455X (gfx1250) — compile-verified
//
#include <hip/hip_runtime.h>

typedef __attribute__((ext_vector_type(2))) float v2f;
typedef __attribute__((ext_vector_type(8))) float v8f;

#define N_SEQ   2048      // B*S
#define T_STEPS 30
#define DIN     500
#define ROWLEN  15000     // T_STEPS * DIN

// ---------------------------------------------------------------------------
// Kernel 1: pre0[t][n][g] = sum_k xf[n,t,k] * w_ih0[g,k] + b_ih0[g] + b_hh0[g]
// via V_WMMA_F32_16X16X4_F32. Flip folded into reversed weights in LDS.
// 8 waves / block, each wave owns 16 rows of the (61440 x 500) A matrix.
// ---------------------------------------------------------------------------
__global__ __launch_bounds__(256) void pre0_wmma_kernel(
    const float* __restrict__ x,     const float* __restrict__ w_ih0,
    const float* __restrict__ b_ih0, const float* __restrict__ b_hh0,
    float* __restrict__ pre)
{
    __shared__ float lds_w[4 * DIN];           // reversed weights, 8 KB
    const int tid = threadIdx.x;
    for (int i = tid; i < 4 * DIN; i += 256) {
        int g = i / DIN;
        int j = i - g * DIN;
        lds_w[i] = w_ih0[g * DIN + (DIN - 1) - j];   // wrev[g][j] = w[g][499-j]
    }
    __syncthreads();

    const int lane = tid & 31;
    const int wave = tid >> 5;
    const int half = lane >> 4;                 // K sub-group (A/B 32-bit layout)
    const int col  = lane & 15;                 // C/D column (gate), A row
    const int rowbase = (blockIdx.x * 8 + wave) * 16;

    // A-matrix row handled by this lane: row r = n*T + t, data chunk is (29-t)
    const int rA = rowbase + col;
    const int nA = rA / T_STEPS;
    const int tA = rA - nA * T_STEPS;
    const float* ap = x + (size_t)nA * ROWLEN + (T_STEPS - 1 - tA) * DIN + 2 * half;

    // B-matrix: gate g lives in column g (cols 4..15 are zero)
    const int wc = (col < 4) ? col : 0;
    const float* bp = &lds_w[wc * DIN + 2 * half];

    v8f acc = {};
    #pragma unroll 5
    for (int k0 = 0; k0 < DIN; k0 += 4) {
        float2 av = *(const float2*)(ap + k0);      // global_load_b64, ascending
        float2 bv = *(const float2*)(bp + k0);      // ds_read_b64
        v2f a; a.x = av.x; a.y = av.y;
        v2f b;
        b.x = (col < 4) ? bv.x : 0.0f;
        b.y = (col < 4) ? bv.y : 0.0f;
        // D = A(16x4) * B(4x16) + C ; f32 throughout
        acc = __builtin_amdgcn_wmma_f32_16x16x4_f32(
            /*neg_a=*/false, a, /*neg_b=*/false, b,
            /*c_mod=*/(short)0, acc, /*reuse_a=*/false, /*reuse_b=*/false);
    }

    // C layout: lane (0-15|16-31), VGPR v -> M = v + 8*half, N = col
    if (col < 4) {
        const float bias = b_ih0[col] + b_hh0[col];
        #pragma unroll
        for (int v = 0; v < 8; ++v) {
            int r = rowbase + v + 8 * half;
            int n = r / T_STEPS;
            int t = r - n * T_STEPS;
            pre[((size_t)t * N_SEQ + n) * 4 + col] = acc[v] + bias;
        }
    }
}

// ---------------------------------------------------------------------------
// Kernel 2: 64 stacked width-1 LSTM layers, one thread per sequence element.
// ---------------------------------------------------------------------------
__device__ __forceinline__ float fsig(float x) {
    float e = __builtin_amdgcn_exp2f(-1.4426950408889634f * x);
    return __builtin_amdgcn_rcpf(1.0f + e);
}
__device__ __forceinline__ float ftanh(float x) {
    float e = __builtin_amdgcn_exp2f(-2.8853900817779268f * x); // 2^(-2x*log2e)
    return 2.0f * __builtin_amdgcn_rcpf(1.0f + e) - 1.0f;
}

__global__ __launch_bounds__(256) void lstm_scan_kernel(
    const float* __restrict__ pre,  const float* __restrict__ w_hh0,
    const float* __restrict__ w_ih, const float* __restrict__ w_hh,
    const float* __restrict__ b_ih, const float* __restrict__ b_hh,
    float* __restrict__ out)
{
    const int n = blockIdx.x * blockDim.x + threadIdx.x;
    if (n >= N_SEQ) return;

    float hbuf[T_STEPS];

    // ---- layer 0: gate preacts come from pre[] (incl. both biases) ----
    {
        const float wh0 = w_hh0[0], wh1 = w_hh0[1], wh2 = w_hh0[2], wh3 = w_hh0[3];
        float h = 0.0f, c = 0.0f;
        #pragma unroll
        for (int t = 0; t < T_STEPS; ++t) {
            const float4 g = *(const float4*)(pre + ((size_t)t * N_SEQ + n) * 4);
            float i  = fsig (g.x + h * wh0);
            float f  = fsig (g.y + h * wh1);
            float cc = ftanh(g.z + h * wh2);
            float o  = fsig (g.w + h * wh3);
            c = f * c + i * cc;
            h = o * ftanh(c);
            hbuf[t] = h;
        }
    }

    // ---- layers 1..63: scalar input = previous layer's hidden stream ----
    for (int l = 0; l < 63; ++l) {
        const float wi0 = w_ih[l * 4 + 0], wi1 = w_ih[l * 4 + 1];
        const float wi2 = w_ih[l * 4 + 2], wi3 = w_ih[l * 4 + 3];
        const float wh0 = w_hh[l * 4 + 0], wh1 = w_hh[l * 4 + 1];
        const float wh2 = w_hh[l * 4 + 2], wh3 = w_hh[l * 4 + 3];
        const float bb0 = b_ih[l * 4 + 0] + b_hh[l * 4 + 0];
        const float bb1 = b_ih[l * 4 + 1] + b_hh[l * 4 + 1];
        const float bb2 = b_ih[l * 4 + 2] + b_hh[l * 4 + 2];
        const float bb3 = b_ih[l * 4 + 3] + b_hh[l * 4 + 3];
        float h = 0.0f, c = 0.0f;
        #pragma unroll
        for (int t = 0; t < T_STEPS; ++t) {
            const float xin = hbuf[t];
            float i  = fsig (xin * wi0 + bb0 + h * wh0);
            float f  = fsig (xin * wi1 + bb1 + h * wh1);
            float cc = ftanh(xin * wi2 + bb2 + h * wh2);
            float o  = fsig (xin * wi3 + bb3 + h * wh3);
            c = f * c + i * cc;
            h = o * ftanh(c);
            hbuf[t] = h;
        }
    }

    #pragma unroll
    for (int t = 0; t < T_STEPS; ++t)
        out[(size_t)n * T_STEPS + t] = hbuf[t];
}

// ---------------------------------------------------------------------------
extern "C" void kernel_launch(void* const* d_in, const int* in_sizes, int n_in,
                              void* d_out, int out_size, void* d_ws, size_t ws_size,
                              hipStream_t stream) {
    const float* x     = (const float*)d_in[0];
    const float* w_ih0 = (const float*)d_in[1];
    const float* w_hh0 = (const float*)d_in[2];
    const float* b_ih0 = (const float*)d_in[3];
    const float* b_hh0 = (const float*)d_in[4];
    const float* w_ih  = (const float*)d_in[5];
    const float* w_hh  = (const float*)d_in[6];
    const float* b_ih  = (const float*)d_in[7];
    const float* b_hh  = (const float*)d_in[8];
    float* out = (float*)d_out;
    float* pre = (float*)d_ws;   // T_STEPS * N_SEQ * 4 floats = 983 KB

    // 61440 rows / (8 waves * 16 rows) = 480 blocks
    pre0_wmma_kernel<<<480, 256, 0, stream>>>(x, w_ih0, b_ih0, b_hh0, pre);
    // 2048 sequences
    lstm_scan_kernel<<<8, 256, 0, stream>>>(pre, w_hh0, w_ih, w_hh, b_ih, b_hh, out);
}